// StarTransformerLayer_75960791597426
// MI455X (gfx1250) — compile-verified
//
#include <hip/hip_runtime.h>
#include <hip/hip_bf16.h>

#define BB 8
#define LL 4096
#define HH 256
#define NHD 8
#define HDD 32

typedef float v2f __attribute__((ext_vector_type(2)));
typedef float v8f __attribute__((ext_vector_type(8)));

// ---------------------------------------------------------------------------
// Y[M x 256] = op( X[M x 256] @ W[256 x 256] + bias [+ R] ), op = optional relu
// FP32 WMMA: V_WMMA_F32_16X16X4_F32, K-loop step 4 (64 steps for K=256).
// block = 128 threads (4 waves). blockIdx.x selects a 16-row M tile; wave w
// computes the 16x64 N-slice [w*64, w*64+64) as 4 16x16 WMMA accumulators.
//
// Fragment layouts (CDNA5 ISA 7.12.2, 32-bit, wave32):
//   A 16x4 : lanes 0-15 rows M=0..15 with K={k,k+1}; lanes 16-31 K={k+2,k+3}
//   B 4x16 : lanes 0-15 cols N=0..15 rows K={k,k+1}; lanes 16-31 K={k+2,k+3}
//   C 16x16: lane<16 -> col N=lane rows 0..7 in v[0..7]; lane>=16 -> rows 8..15
//
// Partial-M tiles (M=8 center GEMMs): A row index is CLAMPED (not predicated)
// so the inner loop is branch-free; D rows >= M contain garbage but the
// epilogue store guard drops them. Reads stay in-bounds.
// ---------------------------------------------------------------------------
__global__ __launch_bounds__(128)
void gemm256(const float* __restrict__ X, const float* __restrict__ W,
             const float* __restrict__ bias, const float* __restrict__ R,
             float* __restrict__ Y, int M, int relu) {
  const int lane = threadIdx.x & 31;
  const int wave = threadIdx.x >> 5;
  const int row0 = blockIdx.x * 16;
  const int n0   = wave * 64;
  const int half = lane >> 4;   // 0: lanes 0-15, 1: lanes 16-31
  const int l16  = lane & 15;

  v8f c[4];
#pragma unroll
  for (int t = 0; t < 4; ++t)
#pragma unroll
    for (int r = 0; r < 8; ++r) c[t][r] = 0.f;

  int arow = row0 + l16;
  if (arow >= M) arow = M - 1;                 // clamp: branch-free inner loop
  const float* __restrict__ xrow = X + (size_t)arow * HH;

  for (int k = 0; k < HH; k += 4) {
    const int kk = k + half * 2;
    v2f a;
    a.x = xrow[kk];
    a.y = xrow[kk + 1];
    const float* __restrict__ wp = W + (size_t)kk * HH + n0 + l16;
#pragma unroll
    for (int t = 0; t < 4; ++t) {
      v2f b;
      b.x = wp[t * 16];
      b.y = wp[HH + t * 16];
      c[t] = __builtin_amdgcn_wmma_f32_16x16x4_f32(
          false, a, false, b, (short)0, c[t], false, false);
    }
  }

#pragma unroll
  for (int t = 0; t < 4; ++t) {
    const int n = n0 + t * 16 + l16;
    const float bb = bias[n];
#pragma unroll
    for (int r = 0; r < 8; ++r) {
      const int row = row0 + half * 8 + r;
      if (row < M) {
        float val = c[t][r] + bb;
        if (R) val += R[(size_t)row * HH + n];
        if (relu) val = fmaxf(val, 0.f);
        Y[(size_t)row * HH + n] = val;
      }
    }
  }
}

// ---------------------------------------------------------------------------
// center = mean over L of x : one thread per (b, h)
// ---------------------------------------------------------------------------
__global__ __launch_bounds__(256)
void center_mean(const float* __restrict__ x, float* __restrict__ center) {
  const int idx = blockIdx.x * 256 + threadIdx.x;   // over B*H = 2048
  const int b = idx / HH, h = idx % HH;
  const float* p = x + (size_t)b * LL * HH + h;
  float s = 0.f;
  for (int i = 0; i < LL; ++i) s += p[(size_t)i * HH];
  center[idx] = s * (1.f / (float)LL);
}

// ---------------------------------------------------------------------------
// sat attention: 5 keys per (token, head). One wave per head, one block per
// token. lane = head dim d. 5 wave-reduced dots -> 5-way softmax -> ctx.
// keys: k0=Kc[i+1 mod L], k1=Kc[i], k2=Kc[i==0?L-1:0], k3=Kx[i], k4=Kcen[b]
// ---------------------------------------------------------------------------
__global__ __launch_bounds__(256)
void sat_attn(const float* __restrict__ Qc, const float* __restrict__ Kc,
              const float* __restrict__ Vc, const float* __restrict__ Kx,
              const float* __restrict__ Vx, const float* __restrict__ Kcen,
              const float* __restrict__ Vcen, float* __restrict__ ctx) {
  const int t = blockIdx.x;               // b*L + i
  const int b = t >> 12;                  // L = 4096
  const int i = t & (LL - 1);
  const int h = threadIdx.x >> 5;
  const int d = threadIdx.x & 31;
  const int col = h * HDD + d;
  const size_t rowT = (size_t)t * HH;
  const size_t base = (size_t)b * LL * HH;

  const int i_before = (i + 1) & (LL - 1);
  const int i_after  = (i == 0) ? (LL - 1) : 0;

  const float q  = Qc[rowT + col];
  const float k0 = Kc[base + (size_t)i_before * HH + col];
  const float k1 = Kc[rowT + col];
  const float k2 = Kc[base + (size_t)i_after * HH + col];
  const float k3 = Kx[rowT + col];
  const float k4 = Kcen[(size_t)b * HH + col];
  const float v0 = Vc[base + (size_t)i_before * HH + col];
  const float v1 = Vc[rowT + col];
  const float v2 = Vc[base + (size_t)i_after * HH + col];
  const float v3 = Vx[rowT + col];
  const float v4 = Vcen[(size_t)b * HH + col];

  float s0 = q * k0, s1 = q * k1, s2 = q * k2, s3 = q * k3, s4 = q * k4;
#pragma unroll
  for (int o = 16; o; o >>= 1) {
    s0 += __shfl_xor(s0, o, 32);
    s1 += __shfl_xor(s1, o, 32);
    s2 += __shfl_xor(s2, o, 32);
    s3 += __shfl_xor(s3, o, 32);
    s4 += __shfl_xor(s4, o, 32);
  }
  const float sc = 0.17677669529663687f;  // 1/sqrt(32)
  s0 *= sc; s1 *= sc; s2 *= sc; s3 *= sc; s4 *= sc;
  const float mx = fmaxf(fmaxf(fmaxf(s0, s1), fmaxf(s2, s3)), s4);
  const float e0 = __expf(s0 - mx), e1 = __expf(s1 - mx), e2 = __expf(s2 - mx),
              e3 = __expf(s3 - mx), e4 = __expf(s4 - mx);
  const float inv = 1.f / (e0 + e1 + e2 + e3 + e4);
  ctx[rowT + col] = (e0 * v0 + e1 * v1 + e2 * v2 + e3 * v3 + e4 * v4) * inv;
}

// ---------------------------------------------------------------------------
// rel attention: one block per (b, h), 4097 keys (key 0 = projected center,
// keys 1..L = projected cur rows). Scores staged in LDS; block softmax.
// ---------------------------------------------------------------------------
__global__ __launch_bounds__(256)
void rel_attn(const float* __restrict__ relQ, const float* __restrict__ kcen,
              const float* __restrict__ relK, const float* __restrict__ vcen,
              const float* __restrict__ relV, float* __restrict__ rctx) {
  const int b = blockIdx.x >> 3;
  const int h = blockIdx.x & 7;
  const int tid = threadIdx.x;
  const int NK = LL + 1;  // 4097

  __shared__ float sc[LL + 1];
  __shared__ float qsh[HDD];
  __shared__ float red[256];

  if (tid < HDD) qsh[tid] = relQ[b * HH + h * HDD + tid];
  __syncthreads();

  for (int j = tid; j < NK; j += 256) {
    const float* kp = (j == 0)
        ? (kcen + (size_t)b * HH + h * HDD)
        : (relK + ((size_t)b * LL + (j - 1)) * HH + h * HDD);
    float s = 0.f;
#pragma unroll
    for (int d2 = 0; d2 < HDD; ++d2) s += qsh[d2] * kp[d2];
    sc[j] = s * 0.17677669529663687f;
  }
  __syncthreads();

  float m = -1e30f;
  for (int j = tid; j < NK; j += 256) m = fmaxf(m, sc[j]);
  red[tid] = m; __syncthreads();
  for (int st = 128; st > 0; st >>= 1) {
    if (tid < st) red[tid] = fmaxf(red[tid], red[tid + st]);
    __syncthreads();
  }
  m = red[0];
  __syncthreads();

  float lsum = 0.f;
  for (int j = tid; j < NK; j += 256) {
    const float e = __expf(sc[j] - m);
    sc[j] = e;
    lsum += e;
  }
  red[tid] = lsum; __syncthreads();
  for (int st = 128; st > 0; st >>= 1) {
    if (tid < st) red[tid] += red[tid + st];
    __syncthreads();
  }
  const float inv = 1.f / red[0];
  __syncthreads();

  const int d = tid & 31, g = tid >> 5;   // 8 groups x 32 dims
  float acc = 0.f;
  for (int j = g; j < NK; j += 8) {
    const float* vp = (j == 0)
        ? (vcen + (size_t)b * HH + h * HDD)
        : (relV + ((size_t)b * LL + (j - 1)) * HH + h * HDD);
    acc += sc[j] * vp[d];
  }
  red[g * 32 + d] = acc; __syncthreads();
  if (tid < 32) {
    float t2 = 0.f;
#pragma unroll
    for (int gg = 0; gg < 8; ++gg) t2 += red[gg * 32 + tid];
    rctx[(size_t)b * HH + h * HDD + tid] = t2 * inv;
  }
}

// ---------------------------------------------------------------------------
// Row LayerNorm over H=256 (input already ReLU'd by the GEMM epilogue).
// One 256-thread block per row.
// ---------------------------------------------------------------------------
__global__ __launch_bounds__(256)
void ln_rows(const float* __restrict__ X, const float* __restrict__ g,
             const float* __restrict__ bt, float* __restrict__ Y) {
  const size_t row = blockIdx.x;
  const int tid = threadIdx.x;
  const float v = X[row * HH + tid];

  __shared__ float red[8];
  float s = v;
#pragma unroll
  for (int o = 16; o; o >>= 1) s += __shfl_xor(s, o, 32);
  if ((tid & 31) == 0) red[tid >> 5] = s;
  __syncthreads();
  float tot = 0.f;
#pragma unroll
  for (int wv = 0; wv < 8; ++wv) tot += red[wv];
  const float mu = tot * (1.f / (float)HH);
  __syncthreads();

  const float d = v - mu;
  float s2 = d * d;
#pragma unroll
  for (int o = 16; o; o >>= 1) s2 += __shfl_xor(s2, o, 32);
  if ((tid & 31) == 0) red[tid >> 5] = s2;
  __syncthreads();
  float tv = 0.f;
#pragma unroll
  for (int wv = 0; wv < 8; ++wv) tv += red[wv];
  const float var = tv * (1.f / (float)HH);

  Y[row * HH + tid] = g[tid] * d * rsqrtf(var + 1e-12f) + bt[tid];
}

// ---------------------------------------------------------------------------
extern "C" void kernel_launch(void* const* d_in, const int* in_sizes, int n_in,
                              void* d_out, int out_size, void* d_ws, size_t ws_size,
                              hipStream_t stream) {
  (void)in_sizes; (void)n_in; (void)out_size; (void)ws_size;

  const float* x      = (const float*)d_in[0];
  const float* satWq  = (const float*)d_in[1];
  const float* satbq  = (const float*)d_in[2];
  const float* satWk  = (const float*)d_in[3];
  const float* satbk  = (const float*)d_in[4];
  const float* satWv  = (const float*)d_in[5];
  const float* satbv  = (const float*)d_in[6];
  const float* satWo  = (const float*)d_in[7];
  const float* satbo  = (const float*)d_in[8];
  const float* relWq  = (const float*)d_in[9];
  const float* relbq  = (const float*)d_in[10];
  const float* relWk  = (const float*)d_in[11];
  const float* relbk  = (const float*)d_in[12];
  const float* relWv  = (const float*)d_in[13];
  const float* relbv  = (const float*)d_in[14];
  const float* relWo  = (const float*)d_in[15];
  const float* relbo  = (const float*)d_in[16];
  const float* satlnw = (const float*)d_in[17];
  const float* satlnb = (const float*)d_in[18];
  const float* rellnw = (const float*)d_in[19];
  const float* rellnb = (const float*)d_in[20];

  const size_t S = (size_t)BB * LL * HH;  // 8,388,608
  const int    M = BB * LL;               // 32768
  const int    MT = M / 16;               // 2048 M tiles

  float* ws   = (float*)d_ws;
  float* curA = ws + 0 * S;
  float* Qb   = ws + 1 * S;   // also reused as pre-LN sat output
  float* Kb   = ws + 2 * S;
  float* Vb   = ws + 3 * S;
  float* Kx   = ws + 4 * S;
  float* Vx   = ws + 5 * S;
  float* Cx   = ws + 6 * S;   // attention context
  float* sm   = ws + 7 * S;   // small buffers
  float* cenA   = sm +  0 * 2048;
  float* cenB   = sm +  1 * 2048;
  float* KcenS  = sm +  2 * 2048;
  float* VcenS  = sm +  3 * 2048;
  float* relQ   = sm +  4 * 2048;
  float* kcen   = sm +  5 * 2048;
  float* vcen   = sm +  6 * 2048;
  float* rctx   = sm +  7 * 2048;
  float* relTmp = sm +  8 * 2048;

  float* outCur = (float*)d_out;
  float* outCen = (float*)d_out + S;

  // ---- precompute (sat K/V projections of x are iteration-invariant) ----
  center_mean<<<BB, 256, 0, stream>>>(x, cenA);
  gemm256<<<MT, 128, 0, stream>>>(x, satWk, satbk, nullptr, Kx, M, 0);
  gemm256<<<MT, 128, 0, stream>>>(x, satWv, satbv, nullptr, Vx, M, 0);

  const float* cur = x;
  const float* cenOld = cenA;
  for (int it = 0; it < 2; ++it) {
    float* curNext = (it == 0) ? curA : outCur;
    float* cenNext = (it == 0) ? cenB : outCen;

    // ---- sat phase ----
    gemm256<<<MT, 128, 0, stream>>>(cur, satWq, satbq, nullptr, Qb, M, 0);
    gemm256<<<MT, 128, 0, stream>>>(cur, satWk, satbk, nullptr, Kb, M, 0);
    gemm256<<<MT, 128, 0, stream>>>(cur, satWv, satbv, nullptr, Vb, M, 0);
    gemm256<<<1, 128, 0, stream>>>(cenOld, satWk, satbk, nullptr, KcenS, BB, 0);
    gemm256<<<1, 128, 0, stream>>>(cenOld, satWv, satbv, nullptr, VcenS, BB, 0);
    sat_attn<<<M, 256, 0, stream>>>(Qb, Kb, Vb, Kx, Vx, KcenS, VcenS, Cx);
    // out = relu(ctx @ Wo + bo + cur) -> Qb (Qb no longer needed)
    gemm256<<<MT, 128, 0, stream>>>(Cx, satWo, satbo, cur, Qb, M, 1);
    ln_rows<<<M, 256, 0, stream>>>(Qb, satlnw, satlnb, curNext);

    // ---- rel phase (queries old center over [center; curNext]) ----
    gemm256<<<1, 128, 0, stream>>>(cenOld, relWq, relbq, nullptr, relQ, BB, 0);
    gemm256<<<1, 128, 0, stream>>>(cenOld, relWk, relbk, nullptr, kcen, BB, 0);
    gemm256<<<1, 128, 0, stream>>>(cenOld, relWv, relbv, nullptr, vcen, BB, 0);
    gemm256<<<MT, 128, 0, stream>>>(curNext, relWk, relbk, nullptr, Kb, M, 0);
    gemm256<<<MT, 128, 0, stream>>>(curNext, relWv, relbv, nullptr, Vb, M, 0);
    rel_attn<<<BB * NHD, 256, 0, stream>>>(relQ, kcen, Kb, vcen, Vb, rctx);
    gemm256<<<1, 128, 0, stream>>>(rctx, relWo, relbo, cenOld, relTmp, BB, 1);
    ln_rows<<<BB, 256, 0, stream>>>(relTmp, rellnw, rellnb, cenNext);

    cur = curNext;
    cenOld = cenNext;
  }
}